// NegativeHardestContrastiveLoss_4672924418560
// MI455X (gfx1250) — compile-verified
//
#include <hip/hip_runtime.h>

typedef float v2f __attribute__((ext_vector_type(2)));
typedef float v4f __attribute__((ext_vector_type(4)));
typedef float v8f __attribute__((ext_vector_type(8)));

#define CC    256          // channels (K)
#define HW    262144       // 512*512 columns (N)
#define NPAIR 64           // M
#define NTILE 128          // columns per block (8 waves * 16)
#define NBLK  (HW / NTILE) // 2048
#define TOPK  5
#define FBIG  3.402823466e+38f
#define NKS   (CC / 4)     // 64 K-steps
#define NFRAG (NKS * 4 * 32) // v2f fragments: ks * mt * lane = 8192 (64 KB)

// ---------------------------------------------------------------------------
// Kernel 1: gather sel = f1[:, idx] directly into WMMA A-fragment layout:
//   afrag[(ks*4 + mt)*32 + lane] = { sel[m][k], sel[m][k+1] }
//   with k = 4*ks + 2*(lane>=16), m = mt*16 + (lane&15)
// (matches ISA 16x4 f32 A layout: VGPR0=K0|K2, VGPR1=K1|K3, M striped 0-15)
// Also computes ||sel_p||^2.
// ---------------------------------------------------------------------------
__global__ void gather_sel_kernel(const float* __restrict__ f1,
                                  const long long* __restrict__ pairs,
                                  float* __restrict__ afrag,
                                  float* __restrict__ normA)
{
    __shared__ float red[CC];
    const int p = blockIdx.x;    // pair (M index)
    const int k = threadIdx.x;   // channel (K index)
    const long long idx = pairs[2 * p];
    const float v = f1[(size_t)k * HW + (size_t)idx];

    const int ks   = k >> 2;
    const int kr   = k & 3;
    const int lane = ((kr >> 1) << 4) + (p & 15); // hi half-wave for K2/K3
    const int mt   = p >> 4;
    afrag[(((ks << 2) + mt) * 32 + lane) * 2 + (kr & 1)] = v;

    red[k] = v * v;
    __syncthreads();
    for (int s = CC / 2; s > 0; s >>= 1) {
        if (k < s) red[k] += red[k + s];
        __syncthreads();
    }
    if (k == 0) normA[p] = red[0];
}

// 5-smallest insertion (t0 <= t1 <= t2 <= t3 <= t4)
__device__ __forceinline__ void ins5(float v, float& t0, float& t1,
                                     float& t2, float& t3, float& t4)
{
    if (v < t4) {
        if (v < t3) { t4 = t3;
            if (v < t2) { t3 = t2;
                if (v < t1) { t2 = t1;
                    if (v < t0) { t1 = t0; t0 = v; } else t1 = v;
                } else t2 = v;
            } else t3 = v;
        } else t4 = v;
    }
}

// ---------------------------------------------------------------------------
// Kernel 2: WMMA fp32 GEMM (64x256 @ 256x128-per-block) -> relu distances ->
// per-block top-5 min per pair.
// ---------------------------------------------------------------------------
__global__ void __launch_bounds__(256)
dist_topk_kernel(const float* __restrict__ f2,
                 const float* __restrict__ afrag_g,
                 const float* __restrict__ normA,
                 float* __restrict__ parts)
{
    extern __shared__ float smem[];
    v2f*   lds_af = (v2f*)smem;                 // NFRAG v2f = 16384 floats
    float* lds_d  = smem + 2 * NFRAG;           // NPAIR*NTILE = 8192 floats
    float* lds_an = lds_d + NPAIR * NTILE;      // NPAIR floats

    const int tid = threadIdx.x;
    {   // cooperative fill of A fragments, 128-bit transfers
        const v4f* src = (const v4f*)afrag_g;
        v4f*       dst = (v4f*)smem;
        #pragma unroll
        for (int i = 0; i < (2 * NFRAG / 4) / 256; ++i)
            dst[tid + i * 256] = src[tid + i * 256];
    }
    if (tid < NPAIR) lds_an[tid] = normA[tid];
    __syncthreads();

    const int wave   = tid >> 5;
    const int lane   = tid & 31;
    const int lane15 = lane & 15;
    const int khalf  = (lane >> 4) << 1;               // 0 (lanes 0-15) / 2 (16-31)
    const int nloc   = wave * 16 + lane15;             // column within block tile
    const size_t ncol = (size_t)blockIdx.x * NTILE + nloc;

    v8f acc[4];
    #pragma unroll
    for (int mt = 0; mt < 4; ++mt) acc[mt] = (v8f){0.f,0.f,0.f,0.f,0.f,0.f,0.f,0.f};
    float bn = 0.0f;

    // 64 K-steps of 4; B layout: VGPR0 = {K0 | K2}, VGPR1 = {K1 | K3}
    #pragma unroll 4
    for (int ks = 0; ks < NKS; ++ks) {
        const int k0 = ks * 4 + khalf;
        const float* bp = f2 + (size_t)k0 * HW + ncol;
        const float b0 = bp[0];
        const float b1 = bp[HW];
        v2f b; b[0] = b0; b[1] = b1;
        bn += b0 * b0 + b1 * b1;
        const int fr = (ks << 2) * 32 + lane;          // fragment slot base
        #pragma unroll
        for (int mt = 0; mt < 4; ++mt) {
            const v2f a = lds_af[fr + mt * 32];        // one aligned ds_load_b64
            acc[mt] = __builtin_amdgcn_wmma_f32_16x16x4_f32(
                false, a, false, b, (short)0, acc[mt], false, false);
        }
    }

    // complete ||f2_n||^2 : lane l and l^16 hold complementary K residues of same n
    const float bnorm = bn + __shfl_xor(bn, 16, 32);

    // d = ||a||^2 + ||b||^2 - 2 dot ; C/D layout: VGPR r -> M = mt*16 + 8*(lane>=16) + r
    #pragma unroll
    for (int mt = 0; mt < 4; ++mt) {
        #pragma unroll
        for (int r = 0; r < 8; ++r) {
            const int m = mt * 16 + ((lane >> 4) << 3) + r;
            float dv = lds_an[m] + bnorm - 2.0f * acc[mt][r];
            dv = fmaxf(dv, 0.0f);
            lds_d[m * NTILE + nloc] = dv;
        }
    }
    __syncthreads();

    // per-block top-5 smallest per pair (partial candidates for global top-5)
    if (tid < NPAIR) {
        float t0 = FBIG, t1 = FBIG, t2 = FBIG, t3 = FBIG, t4 = FBIG;
        const float* row = lds_d + tid * NTILE;
        for (int j = 0; j < NTILE; ++j) ins5(row[j], t0, t1, t2, t3, t4);
        float* o = parts + ((size_t)blockIdx.x * NPAIR + tid) * TOPK;
        o[0] = t0; o[1] = t1; o[2] = t2; o[3] = t3; o[4] = t4;
    }
}

// ---------------------------------------------------------------------------
// Kernel 3: merge block candidates -> mean of 5 smallest -> average over pairs
// ---------------------------------------------------------------------------
__global__ void finalize_kernel(const float* __restrict__ parts,
                                float* __restrict__ out)
{
    __shared__ float red[NPAIR];
    const int m = threadIdx.x;
    float t0 = FBIG, t1 = FBIG, t2 = FBIG, t3 = FBIG, t4 = FBIG;
    for (int blk = 0; blk < NBLK; ++blk) {
        const float* p = parts + ((size_t)blk * NPAIR + m) * TOPK;
        #pragma unroll
        for (int j = 0; j < TOPK; ++j) ins5(p[j], t0, t1, t2, t3, t4);
    }
    red[m] = (t0 + t1 + t2 + t3 + t4) * (1.0f / TOPK);
    __syncthreads();
    for (int s = NPAIR / 2; s > 0; s >>= 1) {
        if (m < s) red[m] += red[m + s];
        __syncthreads();
    }
    if (m == 0) out[0] = red[0] * (1.0f / NPAIR);
}

// ---------------------------------------------------------------------------
extern "C" void kernel_launch(void* const* d_in, const int* in_sizes, int n_in,
                              void* d_out, int out_size, void* d_ws, size_t ws_size,
                              hipStream_t stream)
{
    const float* f1 = (const float*)d_in[0];
    const float* f2 = (const float*)d_in[1];
    const long long* pairs = (const long long*)d_in[2];
    float* out = (float*)d_out;

    float* afrag = (float*)d_ws;          // 2*NFRAG floats (64 KB)
    float* normA = afrag + 2 * NFRAG;     // NPAIR floats
    float* parts = normA + NPAIR;         // NBLK*NPAIR*TOPK floats (~2.6 MB)

    gather_sel_kernel<<<NPAIR, CC, 0, stream>>>(f1, pairs, afrag, normA);

    const size_t smem = (size_t)(2 * NFRAG + NPAIR * NTILE + NPAIR) * sizeof(float);
    dist_topk_kernel<<<NBLK, 256, smem, stream>>>(f2, afrag, normA, parts);

    finalize_kernel<<<1, NPAIR, 0, stream>>>(parts, out);
}